// GCN_67078799229060
// MI455X (gfx1250) — compile-verified
//
#include <hip/hip_runtime.h>

typedef __attribute__((ext_vector_type(2))) float v2f;
typedef __attribute__((ext_vector_type(8))) float v8f;

#define NFEAT 16

__device__ __forceinline__ void atomic_add_f32(float* p, float v) {
    // Native global_atomic_add_f32 (no CAS loop) — safe here: plain HBM, device scope.
    unsafeAtomicAdd(p, v);
}

// ---------------- init: deg=1 (self loop), s1=0, t=0 ----------------
__global__ void k_init(float* __restrict__ deg, float* __restrict__ s1,
                       float* __restrict__ t, int n) {
    int v = blockIdx.x * blockDim.x + threadIdx.x;
    if (v >= n) return;
    deg[v] = 1.0f;
    s1[v]  = 0.0f;
    float4 z = make_float4(0.f, 0.f, 0.f, 0.f);
    float4* tp = (float4*)(t + (size_t)v * NFEAT);
    tp[0] = z; tp[1] = z; tp[2] = z; tp[3] = z;
}

// ---------------- degree on dst ----------------
__global__ void k_degree(const long long* __restrict__ dst,
                         float* __restrict__ deg, int e) {
    int i = blockIdx.x * blockDim.x + threadIdx.x;
    if (i >= e) return;
    atomic_add_f32(&deg[(int)dst[i]], 1.0f);
}

// ---------------- dis = rsqrt(deg) in place (deg >= 1 always) ----------------
__global__ void k_rsqrt(float* __restrict__ deg, int n) {
    int v = blockIdx.x * blockDim.x + threadIdx.x;
    if (v >= n) return;
    deg[v] = rsqrtf(deg[v]);
}

// ---------------- layer-1 scalar aggregation ----------------
__global__ void k_agg1(const long long* __restrict__ src,
                       const long long* __restrict__ dst,
                       const float* __restrict__ x,
                       const float* __restrict__ dis,
                       float* __restrict__ s1, int e) {
    int i = blockIdx.x * blockDim.x + threadIdx.x;
    if (i >= e) return;
    int s = (int)src[i];
    int d = (int)dst[i];
    atomic_add_f32(&s1[d], x[s] * dis[s] * dis[d]);
}

// ---------------- h = relu((s1 + selfloop) * W1 + b1) ----------------
__global__ void k_hidden(const float* __restrict__ x,
                         const float* __restrict__ dis,
                         const float* __restrict__ s1,
                         const float* __restrict__ W1,
                         const float* __restrict__ b1,
                         float* __restrict__ h, int n) {
    int v = blockIdx.x * blockDim.x + threadIdx.x;
    if (v >= n) return;
    float dv = dis[v];
    float st = s1[v] + x[v] * dv * dv;   // add self-loop contribution
    float4* hp = (float4*)(h + (size_t)v * NFEAT);
#pragma unroll
    for (int q = 0; q < 4; ++q) {
        float4 o;
        o.x = fmaxf(0.f, fmaf(st, W1[4 * q + 0], b1[4 * q + 0]));
        o.y = fmaxf(0.f, fmaf(st, W1[4 * q + 1], b1[4 * q + 1]));
        o.z = fmaxf(0.f, fmaf(st, W1[4 * q + 2], b1[4 * q + 2]));
        o.w = fmaxf(0.f, fmaf(st, W1[4 * q + 3], b1[4 * q + 3]));
        hp[q] = o;
    }
}

// ---------------- layer-2 16-wide aggregation ----------------
__global__ void k_agg2(const long long* __restrict__ src,
                       const long long* __restrict__ dst,
                       const float* __restrict__ dis,
                       const float* __restrict__ h,
                       float* __restrict__ t, int e) {
    int i = blockIdx.x * blockDim.x + threadIdx.x;
    if (i >= e) return;
    int s = (int)src[i];
    int d = (int)dst[i];
    float w = dis[s] * dis[d];
    const float4* hp = (const float4*)(h + (size_t)s * NFEAT);
    float* tp = t + (size_t)d * NFEAT;
#pragma unroll
    for (int q = 0; q < 4; ++q) {
        float4 hv = hp[q];
        atomic_add_f32(&tp[4 * q + 0], hv.x * w);
        atomic_add_f32(&tp[4 * q + 1], hv.y * w);
        atomic_add_f32(&tp[4 * q + 2], hv.z * w);
        atomic_add_f32(&tp[4 * q + 3], hv.w * w);
    }
}

// ---------------- finalize: out = (t + h*dis^2) @ W2 + b2 via WMMA f32 16x16x4 ----
// One wave handles 16 nodes. A = [16 nodes x 16 feat] fed as four 16x4 slices,
// B = W2 zero-padded to [16 k x 16 n] fed as four 4x16 slices.
// A layout (ISA 7.12.2): lane = M (lane&15), VGPR0/1 = K pair, lane-half selects K+0/K+2.
// B layout: lane = N, same K striping. C/D: VGPR r = row r (lanes 0-15) / r+8 (16-31).
__global__ void k_out_wmma(const float* __restrict__ t,
                           const float* __restrict__ h,
                           const float* __restrict__ dis,
                           const float* __restrict__ W2,
                           const float* __restrict__ b2,
                           float* __restrict__ out, int n) {
    int lane = threadIdx.x & 31;
    int wave = threadIdx.x >> 5;
    int base = (blockIdx.x * (blockDim.x >> 5) + wave) * 16;
    if (base >= n) return;              // wave-uniform: EXEC all-1s past here

    int half = lane >> 4;               // 0: K+0..1 / rows 0-7 ; 1: K+2..3 / rows 8-15
    int m    = lane & 15;               // A: row index; B/D: column index

    int node_a = base + m;
    if (node_a >= n) node_a = n - 1;    // clamp loads; stores predicated later
    float dv = dis[node_a];
    float d2 = dv * dv;
    const float* tp = t + (size_t)node_a * NFEAT;
    const float* hp = h + (size_t)node_a * NFEAT;

    v8f c = {};
#pragma unroll
    for (int kk = 0; kk < 4; ++kk) {
        int f0 = kk * 4 + half * 2;     // K indices handled by this lane
        v2f a, b;
        a.x = fmaf(hp[f0 + 0], d2, tp[f0 + 0]);   // tt = t + h*dis^2 (self loop)
        a.y = fmaf(hp[f0 + 1], d2, tp[f0 + 1]);
        // B[k][n] = (n < 2) ? W2[k*2 + n] : 0   (W2 row-major [16,2], padded to 16 cols)
        b.x = (m < 2) ? W2[(f0 + 0) * 2 + m] : 0.0f;
        b.y = (m < 2) ? W2[(f0 + 1) * 2 + m] : 0.0f;
        c = __builtin_amdgcn_wmma_f32_16x16x4_f32(
                /*neg_a=*/false, a, /*neg_b=*/false, b,
                /*c_mod=*/(short)0, c, /*reuse_a=*/false, /*reuse_b=*/false);
    }

    if (m < 2) {                        // only output columns 0,1 are real
        float bb = b2[m];
#pragma unroll
        for (int r = 0; r < 8; ++r) {
            int node = base + r + half * 8;
            if (node < n) out[(size_t)node * 2 + m] = c[r] + bb;
        }
    }
}

extern "C" void kernel_launch(void* const* d_in, const int* in_sizes, int n_in,
                              void* d_out, int out_size, void* d_ws, size_t ws_size,
                              hipStream_t stream) {
    const float*     x   = (const float*)d_in[0];
    const long long* ei  = (const long long*)d_in[1];   // int64 [2, E]
    const float*     W1  = (const float*)d_in[2];       // [1,16]
    const float*     b1  = (const float*)d_in[3];       // [16]
    const float*     W2  = (const float*)d_in[4];       // [16,2]
    const float*     b2  = (const float*)d_in[5];       // [2]
    float*           out = (float*)d_out;               // [N,2]

    int n = in_sizes[0];            // 100000 (x is [N,1])
    int e = in_sizes[1] / 2;        // 3200000
    const long long* src = ei;
    const long long* dst = ei + e;

    // workspace: dis[N] | s1[N] | h[16N] | t[16N]  = 34N floats (~13.6 MB)
    float* ws  = (float*)d_ws;
    float* dis = ws;
    float* s1  = ws + (size_t)n;
    float* h   = ws + 2 * (size_t)n;
    float* t   = h + (size_t)n * NFEAT;

    const int B = 256;
    int gn = (n + B - 1) / B;
    int ge = (e + B - 1) / B;

    k_init  <<<gn, B, 0, stream>>>(dis, s1, t, n);
    k_degree<<<ge, B, 0, stream>>>(dst, dis, e);
    k_rsqrt <<<gn, B, 0, stream>>>(dis, n);
    k_agg1  <<<ge, B, 0, stream>>>(src, dst, x, dis, s1, e);
    k_hidden<<<gn, B, 0, stream>>>(x, dis, s1, W1, b1, h, n);
    k_agg2  <<<ge, B, 0, stream>>>(src, dst, dis, h, t, e);

    int tiles  = (n + 15) / 16;            // 16 nodes per wave
    int blocks = (tiles + 7) / 8;          // 8 waves (256 threads) per block
    k_out_wmma<<<blocks, B, 0, stream>>>(t, h, dis, W2, b2, out, n);
}